// JunmaiLayer_3015067042466
// MI455X (gfx1250) — compile-verified
//
#include <hip/hip_runtime.h>
#include <math.h>

// ---------------------------------------------------------------------------
// JunmaiLayer on MI455X (gfx1250)
//
// B=4, N=16, K=16, H=64, O=1.  W (16,16,16,16,16,64) fp32 = 268 MB, streamed
// exactly once -> ~11.5 us floor at 23.3 TB/s.  6 flops/byte on the stream
// (~140 TFLOP/s fp32 needed) -> MACs on the matrix pipe: v_wmma_f32_16x16x4_f32.
//
//   G[z,de,k,c]          = basis table (256 KB, L2-resident)
//   P_ab[zc,o] (per k)   = sum_de G[zc,de] * W[ab,de,k,o]     <- WMMA GEMM
//   h[z,a,o]            += sum_{k,c} G[z,ab,k,c] * P_ab       <- f32 atomics
//   out = silu(h) @ w_fc + b_fc
//
// Grid: 1024 blocks = 256 (a,b) pairs x 4 de-splits (phase 2 is linear in P,
// so partial-P contributions fold correctly through the atomics).  Block =
// 512 threads = 16 waves; wave w owns k=w.  Per iteration each wave issues
// only 3 VMEM ops: 1 global_load_b64 (A frag) + 2 global_load_b128 (B frag,
// non-temporal so the W stream does not evict the basis table from L2).
// The o<->lane binding is permuted as o = 4*lane + t so each lane's four
// B-tile elements are contiguous 16 B (b128-loadable); C columns and the
// final h writes use the same permutation, so the math is unchanged.
// ---------------------------------------------------------------------------

typedef float v2f __attribute__((ext_vector_type(2)));
typedef float v4f __attribute__((ext_vector_type(4)));
typedef float v8f __attribute__((ext_vector_type(8)));

#define ABUF_ELEMS (16 * 256 * 16)   // basis table, fp32
#define H_ELEMS    (4 * 16 * 64)     // h[z][a][o]
#define NSPLIT     4                 // de-range splits per (a,b) block

// Abuf layout: [k][j][lh][r][comp]  (j = de>>2, lh = (de>>1)&1, comp = de&1,
// r = 3*z + c, rows 12..15 zero-padded).  One lane's A fragment = 8 B.
__device__ __forceinline__ int abuf_idx(int k, int de, int r) {
  int j = de >> 2, lh = (de >> 1) & 1, comp = de & 1;
  return ((((k * 64 + j) * 2 + lh) * 16 + r) * 2) + comp;
}

// ---- kernel 1: build basis table -------------------------------------------
__global__ void k_basis(const float* __restrict__ x, float* __restrict__ Abuf) {
  int idx = blockIdx.x * 256 + threadIdx.x;       // 65536 entries
  int k  = idx >> 12;
  int de = (idx >> 4) & 255;
  int r  = idx & 15;
  float val = 0.0f;
  if (r < 12) {
    int z = r / 3, c = r - 3 * z;
    int d = de >> 4, e = de & 15;
    float dx = x[(z * 16 + d) * 3 + 0] - x[(z * 16 + e) * 3 + 0];
    float dy = x[(z * 16 + d) * 3 + 1] - x[(z * 16 + e) * 3 + 1];
    float dw = x[(z * 16 + d) * 3 + 2] - x[(z * 16 + e) * 3 + 2];
    float nsq = dx * dx + dy * dy + dw * dw + 1e-5f;
    float nrm = sqrtf(nsq);
    float diffc = ((c == 0) ? dx : (c == 1) ? dy : dw) / nsq;
    float cutoff = (nrm < 5.0f)
                     ? 0.5f * (cosf(3.14159265358979323846f * nrm * 0.2f) + 1.0f)
                     : 0.0f;
    float start = expf(-5.0f);
    float mean  = start + (1.0f - start) * ((float)k / 15.0f);
    float ib    = 8.0f / (1.0f - start);          // (2/K*(1-start))^-1
    float beta  = ib * ib;
    float t     = expf(-nrm) - mean;
    val = cutoff * expf(-beta * t * t) * diffc;
  }
  Abuf[abuf_idx(k, de, r)] = val;
}

// ---- kernel 2: zero the h accumulator --------------------------------------
__global__ void k_zero(float* __restrict__ hbuf) {
  hbuf[blockIdx.x * 256 + threadIdx.x] = 0.0f;
}

// ---- kernel 3: W-streaming WMMA contraction --------------------------------
__global__ void __launch_bounds__(512, 1)
k_contract(const float* __restrict__ W, const float* __restrict__ Abuf,
           float* __restrict__ hbuf) {
  __shared__ float Pbuf[16 * 4 * 16 * 16];   // wave-private C spill, 64 KB

  const int ab    = blockIdx.x >> 2;     // (a,b) pair -> contiguous 1 MB W block
  const int split = blockIdx.x & (NSPLIT - 1);
  const int tid   = threadIdx.x;
  const int w     = tid >> 5;            // wave id == k
  const int lane  = tid & 31;
  const int ln    = lane & 15;
  const int lh    = lane >> 4;

  const int j0 = split * (64 / NSPLIT);
  const int j1 = j0 + (64 / NSPLIT);

  // B fragment rows: lanes 0-15 -> de rows {0,1} of chunk, lanes 16-31 -> {2,3}.
  // W_ab flat: (de*16 + k)*64 + o ; o = 4*ln + t (permuted binding).
  const float* bbase = W + (size_t)ab * 262144
                         + (size_t)((2 * lh * 16 + w) * 64 + 4 * ln);
  // A fragment: one b64 per iteration.
  const float* abase = Abuf + (size_t)((((w * 64) * 2 + lh) * 16 + ln) * 2);

  v8f acc0 = {}; v8f acc1 = {}; v8f acc2 = {}; v8f acc3 = {};

#pragma unroll 2
  for (int j = j0; j < j1; ++j) {        // de in chunks of 4
    const v4f* bp = (const v4f*)(bbase + (size_t)j * 4096);
    v4f bx = __builtin_nontemporal_load(bp);          // de row kk   , o=4ln..4ln+3
    v4f by = __builtin_nontemporal_load(bp + 256);    // de row kk+1 (+1024 floats)
    v2f A  = *(const v2f*)(abase + (size_t)j * 64);
    v2f B0 = {bx.x, by.x};
    v2f B1 = {bx.y, by.y};
    v2f B2 = {bx.z, by.z};
    v2f B3 = {bx.w, by.w};
    acc0 = __builtin_amdgcn_wmma_f32_16x16x4_f32(false, A, false, B0, (short)0, acc0, false, false);
    acc1 = __builtin_amdgcn_wmma_f32_16x16x4_f32(false, A, false, B1, (short)0, acc1, false, false);
    acc2 = __builtin_amdgcn_wmma_f32_16x16x4_f32(false, A, false, B2, (short)0, acc2, false, false);
    acc3 = __builtin_amdgcn_wmma_f32_16x16x4_f32(false, A, false, B3, (short)0, acc3, false, false);
  }

  // Spill C fragments to wave-private LDS (DS ops in-order per wave).
  float* P = Pbuf + w * (4 * 16 * 16);
#pragma unroll
  for (int v = 0; v < 8; ++v) {
    int row = lh ? (v + 8) : v;          // C layout: lanes0-15 M=v, lanes16-31 M=v+8
    P[0 * 256 + row * 16 + ln] = acc0[v];
    P[1 * 256 + row * 16 + ln] = acc1[v];
    P[2 * 256 + row * 16 + ln] = acc2[v];
    P[3 * 256 + row * 16 + ln] = acc3[v];
  }

  // Phase 2: contrib[z,o] = sum_c G[z,ab,k,c] * P[3z+c, col]; column col=ln
  // maps to o = 4*ln + t.  Lanes 0-15 handle z=0,1; lanes 16-31 handle z=2,3.
  const int a = ab >> 4;
  float g[12];
#pragma unroll
  for (int r = 0; r < 12; ++r) g[r] = Abuf[abuf_idx(w, ab, r)];

#pragma unroll
  for (int t = 0; t < 4; ++t) {
#pragma unroll
    for (int zz = 0; zz < 2; ++zz) {
      int z = lh * 2 + zz;
      float s = g[3 * z + 0] * P[t * 256 + (3 * z + 0) * 16 + ln]
              + g[3 * z + 1] * P[t * 256 + (3 * z + 1) * 16 + ln]
              + g[3 * z + 2] * P[t * 256 + (3 * z + 2) * 16 + ln];
      atomicAdd(&hbuf[(z * 16 + a) * 64 + 4 * ln + t], s);
    }
  }
}

// ---- kernel 4: silu head ---------------------------------------------------
__global__ void k_head(const float* __restrict__ hbuf, const float* __restrict__ wfc,
                       const float* __restrict__ bfc, float* __restrict__ out) {
  int tid = threadIdx.x;                 // 64 = (z,a)
  float s = bfc[0];
  const float* hp = hbuf + tid * 64;
#pragma unroll 4
  for (int o = 0; o < 64; ++o) {
    float v = hp[o];
    s += (v / (1.0f + expf(-v))) * wfc[o];
  }
  out[tid] = s;
}

extern "C" void kernel_launch(void* const* d_in, const int* in_sizes, int n_in,
                              void* d_out, int out_size, void* d_ws, size_t ws_size,
                              hipStream_t stream) {
  const float* x   = (const float*)d_in[0];   // (4,16,3)
  const float* W   = (const float*)d_in[1];   // (16,16,16,16,16,64)
  const float* wfc = (const float*)d_in[2];   // (64,1)
  const float* bfc = (const float*)d_in[3];   // (1,)
  float* Abuf = (float*)d_ws;                 // 256 KB
  float* hbuf = Abuf + ABUF_ELEMS;            // 16 KB

  k_basis<<<256, 256, 0, stream>>>(x, Abuf);
  k_zero<<<H_ELEMS / 256, 256, 0, stream>>>(hbuf);
  k_contract<<<256 * NSPLIT, 512, 0, stream>>>(W, Abuf, hbuf);
  k_head<<<1, 64, 0, stream>>>(hbuf, wfc, bfc, (float*)d_out);
}